// InverseWaveletTransform_57294863729400
// MI455X (gfx1250) — compile-verified
//
#include <hip/hip_runtime.h>
#include <cstdint>

// Inverse 2D Haar DWT (pywt idwt2, db1), f32.
// In : A,H,V,D each (8,32,256,256) f32.  Out: (8,32,512,512) f32.
// Pure streaming (≈0.37 FLOP/byte) -> optimize the data-movement path:
// async global->LDS DMA (ASYNCcnt), double buffering, NT streaming stores.

typedef float f2 __attribute__((ext_vector_type(2)));
typedef float f4 __attribute__((ext_vector_type(4)));

#define NTHREADS    256
#define ROWS        8                       // input rows per tile
#define W_IN        256                     // input width
#define W_OUT       (2 * W_IN)              // 512
#define TILE_FLOATS (ROWS * W_IN)           // 2048 floats per array per tile
#define TILE_BYTES  (TILE_FLOATS * 4)       // 8192 B per array per tile
#define BUF_FLOATS  (4 * TILE_FLOATS)       // all 4 arrays: 8192 floats = 32 KiB
#define BUF_BYTES   (BUF_FLOATS * 4)

// Stage one tile (4 arrays x 8 rows x 256 cols, contiguous per array) into LDS.
// Each lane copies 16 B per instruction; 8 wave-level async ops per wave.
// IOFFSET is added to BOTH the LDS and the global address (ISA 08 §4.4),
// so offset:4096 advances both streams by 256 float4's in lock-step.
__device__ __forceinline__ void issue_tile(uint32_t lbase, uint32_t goff,
                                           uint64_t gA, uint64_t gH,
                                           uint64_t gV, uint64_t gD,
                                           uint32_t tid)
{
    uint32_t voff = goff + (tid << 4);       // global byte offset (GVS mode)
    uint32_t l    = lbase + (tid << 4);      // LDS byte address

    asm volatile("global_load_async_to_lds_b128 %0, %1, %2"             :: "v"(l),                  "v"(voff), "s"(gA) : "memory");
    asm volatile("global_load_async_to_lds_b128 %0, %1, %2 offset:4096" :: "v"(l),                  "v"(voff), "s"(gA) : "memory");
    asm volatile("global_load_async_to_lds_b128 %0, %1, %2"             :: "v"(l + 1u*TILE_BYTES),  "v"(voff), "s"(gH) : "memory");
    asm volatile("global_load_async_to_lds_b128 %0, %1, %2 offset:4096" :: "v"(l + 1u*TILE_BYTES),  "v"(voff), "s"(gH) : "memory");
    asm volatile("global_load_async_to_lds_b128 %0, %1, %2"             :: "v"(l + 2u*TILE_BYTES),  "v"(voff), "s"(gV) : "memory");
    asm volatile("global_load_async_to_lds_b128 %0, %1, %2 offset:4096" :: "v"(l + 2u*TILE_BYTES),  "v"(voff), "s"(gV) : "memory");
    asm volatile("global_load_async_to_lds_b128 %0, %1, %2"             :: "v"(l + 3u*TILE_BYTES),  "v"(voff), "s"(gD) : "memory");
    asm volatile("global_load_async_to_lds_b128 %0, %1, %2 offset:4096" :: "v"(l + 3u*TILE_BYTES),  "v"(voff), "s"(gD) : "memory");
}

__global__ __launch_bounds__(NTHREADS)
void haar_idwt2_async_kernel(const float* __restrict__ A, const float* __restrict__ H,
                             const float* __restrict__ V, const float* __restrict__ D,
                             float* __restrict__ out, int n_tiles)
{
    __shared__ float smem[2 * BUF_FLOATS];   // 64 KiB, double buffered

    const uint32_t tid = threadIdx.x;
    // Generic address of LDS object: low 32 bits == LDS byte offset
    // (flat->LDS mapping truncates to addr[31:0]).
    const uint32_t lds0 = (uint32_t)(uintptr_t)&smem[0];

    int t = (int)blockIdx.x;
    if (t >= n_tiles) return;

    const uint64_t gA = (uint64_t)A, gH = (uint64_t)H;
    const uint64_t gV = (uint64_t)V, gD = (uint64_t)D;

    int sel = 0;
    issue_tile(lds0, (uint32_t)t * (uint32_t)TILE_BYTES, gA, gH, gV, gD, tid);

    while (true) {
        const int tn = t + (int)gridDim.x;

        if (tn < n_tiles) {
            // Prefetch next tile into the other buffer, then wait until only
            // those 8 ops are outstanding: async loads complete in order, so
            // ASYNCcnt <= 8 means tile t's data is fully in LDS.
            issue_tile(lds0 + (uint32_t)((sel ^ 1) * BUF_BYTES),
                       (uint32_t)tn * (uint32_t)TILE_BYTES, gA, gH, gV, gD, tid);
            asm volatile("s_wait_asynccnt 0x8" ::: "memory");
        } else {
            asm volatile("s_wait_asynccnt 0x0" ::: "memory");
        }
        __syncthreads();   // publish LDS writes of all 8 waves

        // ---- compute + store tile t from buffer `sel` ----
        const float* sb = smem + sel * BUF_FLOATS;
        const int b  = t >> 5;               // image index (32 row-blocks/image)
        const int rb = t & 31;               // row-block within image
        float* obase = out + ((size_t)b * (size_t)(W_OUT * W_OUT));

        #pragma unroll
        for (int it = 0; it < 4; ++it) {
            const uint32_t k = (uint32_t)(it * NTHREADS) + tid;  // 0..1023
            const uint32_t r = k >> 7;       // input row in tile (0..7)
            const uint32_t p = k & 127u;     // column pair (0..127)

            const float* rowp = sb + r * W_IN + (p << 1);
            const f2 a = *(const f2*)(rowp);
            const f2 h = *(const f2*)(rowp + 1u * TILE_FLOATS);
            const f2 v = *(const f2*)(rowp + 2u * TILE_FLOATS);
            const f2 d = *(const f2*)(rowp + 3u * TILE_FLOATS);

            const f2 lp = a + h, lm = a - h;
            const f2 mp = v + d, mm = v - d;
            const f2 x00 = (lp + mp) * 0.5f;   // out[2i  ,2j  ]
            const f2 x01 = (lp - mp) * 0.5f;   // out[2i  ,2j+1]
            const f2 x10 = (lm + mm) * 0.5f;   // out[2i+1,2j  ]
            const f2 x11 = (lm - mm) * 0.5f;   // out[2i+1,2j+1]

            const uint32_t orow = (((uint32_t)rb << 3) + r) << 1;  // 2*row_in
            float* po = obase + (size_t)orow * W_OUT + (p << 2);

            const f4 ev = { x00.x, x01.x, x00.y, x01.y };
            const f4 od = { x10.x, x11.x, x10.y, x11.y };
            __builtin_nontemporal_store(ev, (f4*)po);
            __builtin_nontemporal_store(od, (f4*)(po + W_OUT));
        }

        if (tn >= n_tiles) break;
        __syncthreads();   // all waves done reading buf[sel] before reuse
        t = tn;
        sel ^= 1;
    }
}

extern "C" void kernel_launch(void* const* d_in, const int* in_sizes, int n_in,
                              void* d_out, int out_size, void* d_ws, size_t ws_size,
                              hipStream_t stream)
{
    (void)n_in; (void)out_size; (void)d_ws; (void)ws_size;
    const float* A = (const float*)d_in[0];
    const float* H = (const float*)d_in[1];
    const float* V = (const float*)d_in[2];
    const float* D = (const float*)d_in[3];
    float* out = (float*)d_out;

    const int n_tiles = in_sizes[0] / TILE_FLOATS;          // 8192 for (8,32,256,256)
    const int blocks  = n_tiles < 1024 ? n_tiles : 1024;    // 8 pipelined tiles/block

    haar_idwt2_async_kernel<<<dim3(blocks), dim3(NTHREADS), 0, stream>>>(
        A, H, V, D, out, n_tiles);
}